// QuanvolutionFilter_65481071406661
// MI455X (gfx1250) — compile-verified
//
#include <hip/hip_runtime.h>
#include <hip/hip_bf16.h>

// ---------------------------------------------------------------------------
// Quanvolution: per 2x2 patch, 4-qubit circuit -> <Z_w>, w=0..3.
// Circuit (depends only on params) is folded into a fixed 16x16 complex
// unitary V. Initial state is real => amp = [Re(V);Im(V)] (32x16 real) @ psi0.
// Main kernel: f32 WMMA GEMM  Amp(32, Npatch) = M(32,16) @ Psi(16, Npatch),
// then probs = ar^2+ai^2 and sign-weighted sums for <Z_w>.
// ---------------------------------------------------------------------------

typedef __attribute__((ext_vector_type(2))) float v2f;
typedef __attribute__((ext_vector_type(8))) float v8f;

#define BSZ      4096
#define NPATCH   (BSZ * 196)
#define NTILES   (NPATCH / 16)   // 50176, exact

struct c32 { float r, i; };
__device__ __forceinline__ c32 cmul(c32 a, c32 b) { return { a.r*b.r - a.i*b.i, a.r*b.i + a.i*b.r }; }
__device__ __forceinline__ c32 cadd(c32 a, c32 b) { return { a.r + b.r, a.i + b.i }; }

__device__ __forceinline__ void apply_gate(c32* amp, int w, c32 U00, c32 U01, c32 U10, c32 U11) {
    const int bit = 1 << (3 - w);   // wire w is bit (3-w) of the flat index
#pragma unroll
    for (int i = 0; i < 16; ++i) {
        if (i & bit) continue;
        int i1 = i | bit;
        c32 a0 = amp[i], a1 = amp[i1];
        amp[i]  = cadd(cmul(U00, a0), cmul(U01, a1));
        amp[i1] = cadd(cmul(U10, a0), cmul(U11, a1));
    }
}

__device__ __forceinline__ void apply_cnot(c32* amp, int c, int t) {
    const int bc = 1 << (3 - c), bt = 1 << (3 - t);
#pragma unroll
    for (int i = 0; i < 16; ++i) {
        if ((i & bc) && !(i & bt)) {
            int j = i | bt;
            c32 tmp = amp[i]; amp[i] = amp[j]; amp[j] = tmp;
        }
    }
}

// One block, lanes 0..15: lane j evolves basis state e_j through the circuit,
// producing column j of V. Writes M (32x16 row-major f32) to workspace:
// rows 0..15 = Re(V), rows 16..31 = Im(V).
// Uses hardware __cosf/__sinf (args are |p|/2, small range) to avoid the
// multi-thousand-instruction scalarized libm trig expansion.
__global__ void build_unitary_kernel(const float* __restrict__ params, float* __restrict__ M) {
    const int j = threadIdx.x;
    if (j >= 16) return;
    c32 amp[16];
#pragma unroll
    for (int i = 0; i < 16; ++i) amp[i] = { (i == j) ? 1.0f : 0.0f, 0.0f };

#pragma unroll
    for (int l = 0; l < 2; ++l) {
#pragma unroll
        for (int w = 0; w < 4; ++w) {
            const float* p = params + (l * 4 + w) * 3;
            float a = 0.5f * p[0], b = 0.5f * p[1], c = 0.5f * p[2];
            float ca = __cosf(a), sa = __sinf(a);
            float cb = __cosf(b), sb = __sinf(b);
            float cc = __cosf(c), sc = __sinf(c);
            // RZ @ RY
            c32 m00 = {  cb * ca, -sb * ca };
            c32 m01 = { -cb * sa,  sb * sa };
            c32 m10 = {  cb * sa,  sb * sa };
            c32 m11 = {  cb * ca,  sb * ca };
            // RX @ (RZ @ RY);  RX = [[cc, -i sc], [-i sc, cc]]
            c32 rxd = { cc, 0.0f }, rxo = { 0.0f, -sc };
            c32 U00 = cadd(cmul(rxd, m00), cmul(rxo, m10));
            c32 U01 = cadd(cmul(rxd, m01), cmul(rxo, m11));
            c32 U10 = cadd(cmul(rxo, m00), cmul(rxd, m10));
            c32 U11 = cadd(cmul(rxo, m01), cmul(rxd, m11));
            apply_gate(amp, w, U00, U01, U10, U11);
        }
        apply_cnot(amp, 0, 1);
        apply_cnot(amp, 1, 2);
        apply_cnot(amp, 2, 3);
        apply_cnot(amp, 3, 0);
    }
#pragma unroll
    for (int i = 0; i < 16; ++i) {
        M[i * 16 + j]        = amp[i].r;
        M[(i + 16) * 16 + j] = amp[i].i;
    }
}

// Main kernel: each wave handles tiles of 16 patches.
//  A layout (f32 16x4):  V0 = A[row][k0 + (lane<16?0:2)], V1 = +1, row = lane&15
//  B layout (f32 4x16):  V0 = B[k0 + (lane<16?0:2)][col], V1 = +1, col = lane&15
//  D layout (f32 16x16): VGPR j -> row j + (lane<16?0:8), col = lane&15
__global__ __launch_bounds__(256) void quanv_wmma_kernel(const float* __restrict__ x,
                                                         const float* __restrict__ M,
                                                         float* __restrict__ out) {
    const int lane   = threadIdx.x & 31;
    const int wid    = blockIdx.x * (blockDim.x >> 5) + (threadIdx.x >> 5);
    const int nw     = gridDim.x * (blockDim.x >> 5);
    const int row    = lane & 15;
    const int hi     = (lane >> 4) << 1;   // 0 or 2: K sub-index within chunk
    const int hiOff  = (lane >> 4) << 3;   // 0 or 8: D row offset

    // Load A once: real rows (0..15) and imag rows (16..31), 4 K-chunks each.
    v2f Ar[4], Ai[4];
#pragma unroll
    for (int kc = 0; kc < 4; ++kc) {
        const int k0 = 4 * kc + hi;
        Ar[kc].x = M[row * 16 + k0];
        Ar[kc].y = M[row * 16 + k0 + 1];
        Ai[kc].x = M[(row + 16) * 16 + k0];
        Ai[kc].y = M[(row + 16) * 16 + k0 + 1];
    }

    for (int tile = wid; tile < NTILES; tile += nw) {
        const int patch = tile * 16 + row;
        const int img = patch / 196;
        const int pi  = patch % 196;
        const int pr = pi / 14, pc = pi % 14;
        const float* xb = x + img * 784 + pr * 56 + pc * 2;
        const float x0 = xb[0], x1 = xb[1], x2 = xb[28], x3 = xb[29];

        const float cw0 = __cosf(0.5f * x0), sw0 = __sinf(0.5f * x0);
        const float cw1 = __cosf(0.5f * x1), sw1 = __sinf(0.5f * x1);
        const float cw2 = __cosf(0.5f * x2), sw2 = __sinf(0.5f * x2);
        const float cw3 = __cosf(0.5f * x3), sw3 = __sinf(0.5f * x3);

        // psi[k] = p01[k>>2] * p23[k&3]; this lane only needs p23[hi], p23[hi+1]
        float p01[4] = { cw0 * cw1, cw0 * sw1, sw0 * cw1, sw0 * sw1 };
        const float t2 = (lane >= 16) ? sw2 : cw2;   // p23[hi]   = t2*cw3
        const float p23h0 = t2 * cw3;                // p23[hi+1] = t2*sw3
        const float p23h1 = t2 * sw3;

        v2f B[4];
#pragma unroll
        for (int kc = 0; kc < 4; ++kc) { B[kc].x = p01[kc] * p23h0; B[kc].y = p01[kc] * p23h1; }

        v8f dr = {}; v8f di = {};
#pragma unroll
        for (int kc = 0; kc < 4; ++kc) {
            dr = __builtin_amdgcn_wmma_f32_16x16x4_f32(false, Ar[kc], false, B[kc],
                                                       (short)0, dr, false, false);
            di = __builtin_amdgcn_wmma_f32_16x16x4_f32(false, Ai[kc], false, B[kc],
                                                       (short)0, di, false, false);
        }

        // probs and sign-weighted partial sums (this lane holds rows hiOff..hiOff+7)
        float e0 = 0.f, e1 = 0.f, e2 = 0.f, e3 = 0.f;
#pragma unroll
        for (int j = 0; j < 8; ++j) {
            const float pj = dr[j] * dr[j] + di[j] * di[j];
            const int idx = j + hiOff;
            e0 += (idx & 8) ? -pj : pj;   // wire 0 = bit 3
            e1 += (idx & 4) ? -pj : pj;   // wire 1 = bit 2
            e2 += (idx & 2) ? -pj : pj;   // wire 2 = bit 1
            e3 += (idx & 1) ? -pj : pj;   // wire 3 = bit 0
        }
        // combine the two half-columns (lane L and L^16 hold the same patch)
        e0 += __shfl_xor(e0, 16, 32);
        e1 += __shfl_xor(e1, 16, 32);
        e2 += __shfl_xor(e2, 16, 32);
        e3 += __shfl_xor(e3, 16, 32);

        if (lane < 16) {
            float4 r = make_float4(e0, e1, e2, e3);
            *reinterpret_cast<float4*>(out + patch * 4) = r;
        }
    }
}

extern "C" void kernel_launch(void* const* d_in, const int* in_sizes, int n_in,
                              void* d_out, int out_size, void* d_ws, size_t ws_size,
                              hipStream_t stream) {
    const float* x      = (const float*)d_in[0];   // (4096, 28, 28) f32
    const float* params = (const float*)d_in[1];   // (2, 4, 3) f32
    float* out = (float*)d_out;                    // (4096, 784) f32
    float* M   = (float*)d_ws;                     // 32x16 f32 = 2 KB

    hipLaunchKernelGGL(build_unitary_kernel, dim3(1), dim3(32), 0, stream, params, M);
    // 784 blocks x 8 waves = 6272 waves; 50176 tiles -> exactly 8 tiles/wave
    hipLaunchKernelGGL(quanv_wmma_kernel, dim3(784), dim3(256), 0, stream, x, M, out);
}